// LabelSmoothingLossMoECombined_27367531610373
// MI455X (gfx1250) — compile-verified
//
#include <hip/hip_runtime.h>
#include <hip/hip_bf16.h>
#include <math.h>

typedef __attribute__((ext_vector_type(2))) float v2f;
typedef __attribute__((ext_vector_type(4))) float v4f;
typedef __attribute__((ext_vector_type(8))) float v8f;

#define WAVE 32

// ---------------------------------------------------------------------------
// Wave-wide sum using V_WMMA_F32_16X16X4_F32.
// A (16x4 f32) holds 2 values per lane: A[m][k] with lane = m + 16*(k>=2),
// vgpr = k%2 (doc layout). We place (v, 0) per lane, B = all ones, C = 0.
// Then D[m][n] = rowsum_m(A) for every n. A lane holds D[m][n] for m = vgpr
// (+8 if lane>=16) at n = lane%16, so sum of its 8 regs covers half the m
// range; adding the xor-16 partner (same n, complementary m) gives the total
// sum of all 64 A entries = wave-wide sum. Requires full EXEC at the call.
// ---------------------------------------------------------------------------
__device__ __forceinline__ float wave_sum_wmma(float v) {
    v2f a; a[0] = v;    a[1] = 0.0f;
    v2f b; b[0] = 1.0f; b[1] = 1.0f;
    v8f c = {};
    c = __builtin_amdgcn_wmma_f32_16x16x4_f32(
        /*neg_a=*/false, a, /*neg_b=*/false, b,
        /*c_mod=*/(short)0, c, /*reuse_a=*/false, /*reuse_b=*/false);
    float s = ((c[0] + c[1]) + (c[2] + c[3])) + ((c[4] + c[5]) + (c[6] + c[7]));
    s += __shfl_xor(s, 16, WAVE);
    return s;
}

__device__ __forceinline__ float wave_max(float v) {
    #pragma unroll
    for (int off = 16; off > 0; off >>= 1)
        v = fmaxf(v, __shfl_xor(v, off, WAVE));
    return v;
}

// Online-softmax update for one float4 chunk (amortized: 5 exps / 4 elems).
__device__ __forceinline__ void osm_update(float& m, float& s, float& sx, v4f v) {
    float m4 = fmaxf(fmaxf(v[0], v[1]), fmaxf(v[2], v[3]));
    float nm = fmaxf(m, m4);
    s = s * __expf(m - nm)
      + ((__expf(v[0] - nm) + __expf(v[1] - nm))
       + (__expf(v[2] - nm) + __expf(v[3] - nm)));
    m = nm;
    sx += (v[0] + v[1]) + (v[2] + v[3]);
}

// ---------------------------------------------------------------------------
// Label-smoothing KL: one wave32 per row of x (N rows, V classes).
// Single non-temporal streaming pass (x is read exactly once and is larger
// than L2, so TH=NT is the right policy). Two independent online-softmax
// accumulator pairs to double ILP across the serial (m,s) chain.
// ---------------------------------------------------------------------------
__global__ void ls_rows_kernel(const float* __restrict__ x,
                               const int* __restrict__ target,
                               float* __restrict__ acc, int N, int V,
                               float ent, float smf, float cf) {
    const int waveId = threadIdx.x >> 5;
    const int lane   = threadIdx.x & 31;
    const int row    = blockIdx.x * (blockDim.x >> 5) + waveId;
    if (row >= N) return;                 // wave-uniform exit

    const float* xr = x + (size_t)row * (size_t)V;
    float m = -INFINITY, s = 0.0f, sumx = 0.0f;

    if ((V & 3) == 0) {
        const v4f* xv = reinterpret_cast<const v4f*>(xr);
        const int nvec = V >> 2;
        float m0 = -INFINITY, s0 = 0.0f, sx0 = 0.0f;
        float m1 = -INFINITY, s1 = 0.0f, sx1 = 0.0f;
        int i = lane;
        for (; i + WAVE < nvec; i += 2 * WAVE) {
            v4f v0 = __builtin_nontemporal_load(&xv[i]);
            v4f v1 = __builtin_nontemporal_load(&xv[i + WAVE]);
            osm_update(m0, s0, sx0, v0);
            osm_update(m1, s1, sx1, v1);
        }
        for (; i < nvec; i += WAVE) {
            v4f v0 = __builtin_nontemporal_load(&xv[i]);
            osm_update(m0, s0, sx0, v0);
        }
        m = fmaxf(m0, m1);
        s = s0 * __expf(m0 - m) + s1 * __expf(m1 - m);
        sumx = sx0 + sx1;
    } else {
        for (int i = lane; i < V; i += WAVE) {
            float v  = xr[i];
            float nm = fmaxf(m, v);
            s = s * __expf(m - nm) + __expf(v - nm);
            m = nm;
            sumx += v;
        }
    }

    // wave reductions (reconverged here; full EXEC for WMMA)
    float mw = wave_max(m);
    float sw = wave_sum_wmma(s * __expf(m - mw));   // exp(-inf)=0 safe
    float sx = wave_sum_wmma(sumx);

    if (lane == 0) {
        int  tgt = target[row];
        bool ign = (tgt < 0);
        int  t   = ign ? 0 : tgt;
        float xt = xr[t];
        float lse      = mw + __logf(sw);
        float logp_tgt = xt - lse;
        float sum_logp = sx - (float)V * lse;
        float cross = smf * (sum_logp - logp_tgt) + cf * logp_tgt;
        float kl    = ign ? 0.0f : (ent - cross);
        atomicAdd(acc, kl);
    }
}

// ---------------------------------------------------------------------------
// MoE load-balance: segment-sum of ones and values over experts, then
// E/num_tokens * sum(load_e * sum_e).  Single block (n = N*K is tiny).
// ---------------------------------------------------------------------------
__global__ void lb_kernel(const float* __restrict__ vals,
                          const int* __restrict__ idx,
                          float* __restrict__ acc_slot,
                          int n, int E, float scale) {
    __shared__ float loads[64];
    __shared__ float sums[64];
    for (int e = threadIdx.x; e < E; e += blockDim.x) { loads[e] = 0.0f; sums[e] = 0.0f; }
    __syncthreads();
    for (int i = threadIdx.x; i < n; i += blockDim.x) {
        int e = idx[i];
        atomicAdd(&loads[e], 1.0f);
        atomicAdd(&sums[e],  vals[i]);
    }
    __syncthreads();
    if (threadIdx.x == 0) {
        float r = 0.0f;
        for (int e = 0; e < E; ++e) r += loads[e] * sums[e];
        *acc_slot = r * scale;
    }
}

// ---------------------------------------------------------------------------
// z-loss: sum over rows of logsumexp(gate_logits_row)^2 (mean applied later).
// One thread per row (E small); WMMA wave-sum then one atomic per wave.
// ---------------------------------------------------------------------------
__global__ void z_kernel(const float* __restrict__ gl,
                         float* __restrict__ acc, int N, int E) {
    int i = blockIdx.x * blockDim.x + threadIdx.x;
    float q = 0.0f;
    if (i < N) {
        const float* g = gl + (size_t)i * (size_t)E;
        float m = -INFINITY;
        for (int e = 0; e < E; ++e) m = fmaxf(m, g[e]);
        float s = 0.0f;
        for (int e = 0; e < E; ++e) s += __expf(g[e] - m);
        float lse = m + __logf(s);
        q = lse * lse;
    }
    float ws = wave_sum_wmma(q);   // reconverged: full EXEC
    if ((threadIdx.x & 31) == 0) atomicAdd(acc, ws);
}

__global__ void init_kernel(float* acc) {
    if (threadIdx.x < 8) acc[threadIdx.x] = 0.0f;
}

__global__ void combine_kernel(const float* __restrict__ acc,
                               float* __restrict__ out,
                               float invB, float invN) {
    if (threadIdx.x == 0 && blockIdx.x == 0) {
        float ls = acc[0] * invB;
        float lb = 0.01f  * 0.5f * (acc[1] + acc[2]);
        float z  = 0.001f * 0.5f * (acc[3] + acc[4]) * invN;
        out[0] = ls + lb + z;
    }
}

extern "C" void kernel_launch(void* const* d_in, const int* in_sizes, int n_in,
                              void* d_out, int out_size, void* d_ws, size_t ws_size,
                              hipStream_t stream) {
    const float* x     = (const float*)d_in[0];
    const int*   tgt   = (const int*)  d_in[1];
    const float* tv_l  = (const float*)d_in[2];
    const int*   ti_l  = (const int*)  d_in[3];
    const float* gl_l  = (const float*)d_in[4];
    const float* tv_g  = (const float*)d_in[5];
    const int*   ti_g  = (const int*)  d_in[6];
    const float* gl_g  = (const float*)d_in[7];

    float* acc = (float*)d_ws;

    const int N  = in_sizes[1];             // 8192
    const int V  = in_sizes[0] / N;         // 10000
    const int nk = in_sizes[2];             // N*K = 16384
    const int K  = nk / N;                  // 2
    const int E  = in_sizes[4] / N;         // 8
    const int B  = 8;                       // x.shape[0] per reference

    // Host-side constants (double precision, matching np.log usage).
    const double SMOOTH = 0.1;
    const double smo    = SMOOTH / (double)(V - 1);
    const double CONF   = 1.0 - SMOOTH;
    const float  ent    = (float)((double)(V - 1) * smo * log(smo) + CONF * log(CONF));
    const float  smf    = (float)smo;
    const float  cf     = (float)CONF;

    init_kernel<<<1, 32, 0, stream>>>(acc);

    // one wave per row, 8 rows (256 threads) per block
    const int rowsPerBlock = 8;
    int grid = (N + rowsPerBlock - 1) / rowsPerBlock;
    ls_rows_kernel<<<grid, 256, 0, stream>>>(x, tgt, acc + 0, N, V, ent, smf, cf);

    float lbscale = (float)E * (float)K / (float)nk;   // E / num_tokens
    lb_kernel<<<1, 256, 0, stream>>>(tv_l, ti_l, acc + 1, nk, E, lbscale);
    lb_kernel<<<1, 256, 0, stream>>>(tv_g, ti_g, acc + 2, nk, E, lbscale);

    int zgrid = (N + 255) / 256;
    z_kernel<<<zgrid, 256, 0, stream>>>(gl_l, acc + 3, N, E);
    z_kernel<<<zgrid, 256, 0, stream>>>(gl_g, acc + 4, N, E);

    combine_kernel<<<1, 32, 0, stream>>>(acc, (float*)d_out,
                                         1.0f / (float)B, 1.0f / (float)N);
}